// LossFunc_412316860480
// MI455X (gfx1250) — compile-verified
//
#include <hip/hip_runtime.h>
#include <stdint.h>

typedef float v2f __attribute__((ext_vector_type(2)));
typedef float v8f __attribute__((ext_vector_type(8)));
typedef unsigned int v4u __attribute__((ext_vector_type(4)));
typedef int v4i __attribute__((ext_vector_type(4)));
typedef int v8i __attribute__((ext_vector_type(8)));

#define BETA      0.1f
#define W_CLS     1.0f
#define B_SZ      32
#define N_PTS     2048
#define D_K       4
#define C_CH      4
#define L_LAT     256
#define NT        128            // N_PTS / 16 tiles
#define THREADS_A 512
#define WAVES_A   16
#define TPW       8              // tiles per wave = NT / WAVES_A

// monotone float->uint mapping (order-preserving incl. negatives)
__device__ __forceinline__ uint32_t f32_key(float f) {
    uint32_t b = __float_as_uint(f);
    uint32_t mask = 0x80000000u | (uint32_t)((int32_t)b >> 31);
    return b ^ mask;
}
__device__ __forceinline__ float key_f32(uint32_t u) {
    uint32_t b = (u & 0x80000000u) ? (u ^ 0x80000000u) : ~u;
    return __uint_as_float(b);
}

// TDM: stage `bytes/4` f32 elements (one contiguous row) from global -> LDS.
// D# group0: count=1, lds_addr, 57-bit global addr, type=2 ("image").
// D# group1: data_size=4B, tensor_dim0=tile_dim0=nelem, dims1=1, stride=nelem.
__device__ __forceinline__ void tdm_load_row_f32(uint32_t lds_off,
                                                 const float* gptr,
                                                 uint32_t nelem) {
    uint64_t ga = (uint64_t)(uintptr_t)gptr;
    v4u g0;
    g0.x = 1u;                                     // count=1, user descriptor
    g0.y = lds_off;                                // lds_addr (bytes)
    g0.z = (uint32_t)ga;                           // global_addr[31:0]
    g0.w = (uint32_t)((ga >> 32) & 0x01FFFFFFu)    // global_addr[56:32]
         | 0x80000000u;                            // type=2 in bits [127:126]
    v8i g1;
    g1[0] = 0x00020000;                            // data_size=2 -> 4 bytes
    g1[1] = (int)((nelem & 0xFFFFu) << 16);        // tensor_dim0[15:0]
    g1[2] = (int)(((nelem >> 16) & 0xFFFFu)        // tensor_dim0[31:16]
         | (1u << 16));                            // tensor_dim1 = 1
    g1[3] = (int)((nelem & 0xFFFFu) << 16);        // tile_dim0 (16-bit field)
    g1[4] = 1;                                     // tile_dim1=1, tile_dim2=0
    g1[5] = (int)nelem;                            // tensor_dim0_stride[31:0]
    g1[6] = 0;
    g1[7] = 0;
    v4i gz = {0, 0, 0, 0};
#if defined(__clang_major__) && __clang_major__ >= 23
    v8i gz8 = {0, 0, 0, 0, 0, 0, 0, 0};
    __builtin_amdgcn_tensor_load_to_lds(g0, g1, gz, gz, gz8, 0);
#else
    __builtin_amdgcn_tensor_load_to_lds(g0, g1, gz, gz, 0);
#endif
}

// One workgroup per batch. Y batch (32KB) staged to LDS via the Tensor Data
// Mover once; y^2 table precomputed in LDS. 16 waves, each owns 8 x-row tiles
// and sweeps all 128 y-col tiles with V_WMMA_F32_16X16X4_F32 computing
//   D = (-2x)·y + (x^2 + y^2)   (norms folded into the C accumulator).
// dist1/idx1: packed u64 keys in registers + shfl butterfly, processed inline.
// dist2/idx2: ds_min_u64 atomics into an LDS table.
__global__ __launch_bounds__(THREADS_A) void chamfer_class_kernel(
    const float* __restrict__ kine_input, const float* __restrict__ class_input,
    const float* __restrict__ kine_pred,  const float* __restrict__ class_pred,
    float* __restrict__ partial) {
  __shared__ float sy[N_PTS * D_K];             // 32KB staged kine_pred batch
  __shared__ float sy2[N_PTS];                  // 8KB  |y|^2 table
  __shared__ unsigned long long skey2[N_PTS];   // 16KB per-m (dist2,idx2)

  const int b    = blockIdx.x;
  const int tid  = threadIdx.x;
  const int lane = tid & 31;
  const int wave = tid >> 5;
  const int col  = lane & 15;   // column within tile
  const int hh   = lane >> 4;   // lane half: comps 0-1 vs 2-3 / rows 0-7 vs 8-15

  const float* Xb  = kine_input  + (size_t)b * N_PTS * D_K;
  const float* Yb  = kine_pred   + (size_t)b * N_PTS * D_K;
  const float* CIb = class_input + (size_t)b * N_PTS * C_CH;
  const float* CPb = class_pred  + (size_t)b * N_PTS * C_CH;

  for (int i = tid; i < N_PTS; i += THREADS_A) skey2[i] = ~0ull;

  if (wave == 0) {  // one wave issues the TDM DMA for the whole block
    tdm_load_row_f32((uint32_t)(uintptr_t)(void*)sy, Yb, N_PTS * D_K);
    __builtin_amdgcn_s_wait_tensorcnt(0);
  }
  __syncthreads();

  // |y|^2 table
  for (int p = tid; p < N_PTS; p += THREADS_A) {
    float a = sy[p * 4 + 0], bb = sy[p * 4 + 1];
    float c = sy[p * 4 + 2], dd = sy[p * 4 + 3];
    sy2[p] = a * a + bb * bb + c * c + dd * dd;
  }
  __syncthreads();

  float acc = 0.0f;

  for (int txi = 0; txi < TPW; ++txi) {
    const int xt = wave * TPW + txi;
    // A-matrix 16x4 f32 layout: lanes 0-15 hold K=0,1 of row M=lane,
    // lanes 16-31 hold K=2,3 of row M=lane-16  (D=4 contiguous in memory)
    const float* xp = Xb + ((xt * 16 + col) * D_K + 2 * hh);
    float ax = xp[0], ay = xp[1];
    float part = ax * ax + ay * ay;
    float x2 = part + __shfl_xor(part, 16, 32);   // |x|^2 of point xt*16+col
    float xn2[8];
#pragma unroll
    for (int r = 0; r < 8; ++r)
      xn2[r] = __shfl(x2, r + 8 * hh, 32);        // |x|^2 of my 8 C-rows
    v2f amat; amat.x = -2.0f * ax; amat.y = -2.0f * ay;

    unsigned long long key1r[8];
#pragma unroll
    for (int r = 0; r < 8; ++r) key1r[r] = ~0ull;

    for (int yt = 0; yt < NT; ++yt) {
      const int m0 = yt * 16 + col;
      v2f bmat = *(const v2f*)&sy[m0 * D_K + 2 * hh];   // ds_load_b64
      float y2col = sy2[m0];                            // ds_load_b32
      v8f c;
#pragma unroll
      for (int r = 0; r < 8; ++r) c[r] = xn2[r] + y2col;

      // dist tile = x^2 + y^2 - 2 x.y directly out of the WMMA
      v8f d = __builtin_amdgcn_wmma_f32_16x16x4_f32(
          false, amat, false, bmat, (short)0, c, false, false);

      unsigned long long km = ~0ull;
#pragma unroll
      for (int r = 0; r < 8; ++r) {
        uint32_t t = f32_key(d[r]);
        unsigned long long k1 = ((unsigned long long)t << 32) | (uint32_t)m0;
        if (k1 < key1r[r]) key1r[r] = k1;
        unsigned long long k2 = ((unsigned long long)t << 32) |
                                (uint32_t)(xt * 16 + r + 8 * hh);
        if (k2 < km) km = k2;
      }
      atomicMin(&skey2[m0], km);                        // ds_min_u64
    }

    // reduce key1r over the 16 columns of each lane-half
#pragma unroll
    for (int m = 1; m <= 8; m <<= 1) {
#pragma unroll
      for (int r = 0; r < 8; ++r) {
        unsigned long long o = __shfl_xor(key1r[r], m, 32);
        if (o < key1r[r]) key1r[r] = o;
      }
    }
    // lanes 0 and 16 finish dist1 + matched-class gather for their 8 rows
    if (col == 0) {
#pragma unroll
      for (int r = 0; r < 8; ++r) {
        const int n = xt * 16 + 8 * hh + r;
        unsigned long long k1 = key1r[r];
        acc += key_f32((uint32_t)(k1 >> 32));           // dist1
        const float* sp = CPb + (int)(uint32_t)k1 * C_CH;  // class_pred[idx1]
        const float* ci = CIb + n * C_CH;
        acc += W_CLS * (sp[0]*ci[0] + sp[1]*ci[1] + sp[2]*ci[2] + sp[3]*ci[3]);
      }
    }
  }
  __syncthreads();

  // Phase 2: dist2 + matched-class gather per pred point m
  for (int m = tid; m < N_PTS; m += THREADS_A) {
    unsigned long long k2 = skey2[m];
    acc += key_f32((uint32_t)(k2 >> 32));               // dist2
    const float* si = CIb + (int)(uint32_t)k2 * C_CH;   // class_input[idx2]
    const float* cp = CPb + m * C_CH;
    acc += W_CLS * (si[0]*cp[0] + si[1]*cp[1] + si[2]*cp[2] + si[3]*cp[3]);
  }
  __syncthreads();

  // block reduction (reuse skey2 region as float scratch)
  float* sred = (float*)skey2;
  sred[tid] = acc;
  __syncthreads();
  for (int s = THREADS_A / 2; s > 0; s >>= 1) {
    if (tid < s) sred[tid] += sred[tid + s];
    __syncthreads();
  }
  if (tid == 0) partial[b] = sred[0];
}

__global__ __launch_bounds__(256) void kl_kernel(
    const float* __restrict__ mu, const float* __restrict__ log_var,
    float* __restrict__ partial) {
  __shared__ float sred[256];
  const int tid = threadIdx.x;
  float acc = 0.0f;
  for (int i = tid; i < B_SZ * L_LAT; i += 256) {
    float m = mu[i], lv = log_var[i];
    acc += 1.0f + lv - m * m - expf(lv);
  }
  sred[tid] = acc;
  __syncthreads();
  for (int s = 128; s > 0; s >>= 1) {
    if (tid < s) sred[tid] += sred[tid + s];
    __syncthreads();
  }
  if (tid == 0) partial[B_SZ] = -0.5f * sred[0];
}

__global__ void finalize_kernel(const float* __restrict__ partial,
                                float* __restrict__ out) {
  if (threadIdx.x == 0 && blockIdx.x == 0) {
    float s = 0.0f;
    for (int i = 0; i < B_SZ; ++i) s += partial[i];   // fixed order: deterministic
    out[0] = (1.0f - BETA) * s + BETA * partial[B_SZ];
  }
}

extern "C" void kernel_launch(void* const* d_in, const int* in_sizes, int n_in,
                              void* d_out, int out_size, void* d_ws, size_t ws_size,
                              hipStream_t stream) {
  const float* kine_input  = (const float*)d_in[0];
  const float* class_input = (const float*)d_in[1];
  const float* kine_pred   = (const float*)d_in[2];
  const float* class_pred  = (const float*)d_in[3];
  const float* mu          = (const float*)d_in[4];
  const float* log_var     = (const float*)d_in[5];
  float* partial = (float*)d_ws;      // [0..31] per-batch chamfer+class, [32] KL
  float* out = (float*)d_out;

  chamfer_class_kernel<<<B_SZ, THREADS_A, 0, stream>>>(
      kine_input, class_input, kine_pred, class_pred, partial);
  kl_kernel<<<1, 256, 0, stream>>>(mu, log_var, partial);
  finalize_kernel<<<1, 64, 0, stream>>>(partial, out);
}